// Method_GCN_81243601371888
// MI455X (gfx1250) — compile-verified
//
#include <hip/hip_runtime.h>
#include <hip/hip_bf16.h>
#include <math.h>

typedef __attribute__((ext_vector_type(2))) float v2f;
typedef __attribute__((ext_vector_type(8))) float v8f;

#define F_IN  500
#define F_HID 16
#define F_OUT 3

// ---------------- init: deg=1 (self loop), zero accumulators ----------------
__global__ void init_kernel(float* __restrict__ deg, float* __restrict__ agg1,
                            float* __restrict__ agg2, int n) {
    int i = blockIdx.x * blockDim.x + threadIdx.x;
    if (i < n) deg[i] = 1.0f;                 // self-loop contribution to degree
    if (i < n * F_OUT) agg2[i] = 0.0f;
    if (i < n * F_HID) agg1[i] = 0.0f;
}

// ---------------- degree scatter over real edges ----------------
__global__ void degree_kernel(const int* __restrict__ dst, int E, float* __restrict__ deg) {
    int e = blockIdx.x * blockDim.x + threadIdx.x;
    if (e < E) atomicAdd(&deg[dst[e]], 1.0f);
}

// ---------------- deg -> deg^{-1/2} in place ----------------
__global__ void rsqrt_kernel(float* __restrict__ deg, int n) {
    int i = blockIdx.x * blockDim.x + threadIdx.x;
    if (i < n) deg[i] = rsqrtf(deg[i]);       // deg >= 1 always (self-loops)
}

// ---------------- h1 = x @ W1 via V_WMMA_F32_16X16X4_F32 ----------------
// One wave32 per 16-row tile; N=16 covers the full hidden width.
// A (16x4 f32): lane l (l16=l&15, half=l>>4): VGPR0 = x[row=l16][k+2*half],
//               VGPR1 = x[row=l16][k+2*half+1]   (8B contiguous NT load)
// B (4x16 f32): VGPR0 = W1[k+2*half][n=l16], VGPR1 = W1[k+2*half+1][n=l16]
// C/D (16x16):  VGPR r -> row (r + 8*half), col l16
__global__ void gemm1_wmma_kernel(const float* __restrict__ x,
                                  const float* __restrict__ W1,
                                  float* __restrict__ h1, int n, int ntiles) {
    int wave = (blockIdx.x * blockDim.x + threadIdx.x) >> 5;
    int lane = threadIdx.x & 31;
    if (wave >= ntiles) return;               // wave-uniform exit: EXEC stays all-1s
    int row0 = wave * 16;
    int half = lane >> 4;
    int l16  = lane & 15;
    int row  = row0 + l16;
    if (row >= n) row = n - 1;                // clamp (keeps all lanes active)
    const float* arow = x + (size_t)row * F_IN;

    v8f c0 = {}; v8f c1 = {};
    int k = 0;
    for (; k + 8 <= F_IN; k += 8) {           // 2 accumulators break the WMMA RAW chain
        int ka0 = k + 2 * half;
        int ka1 = k + 4 + 2 * half;
        // x is 200MB read-once (> 192MB L2): non-temporal so it doesn't evict
        // the L2-resident edge list / h buffers used by later kernels.
        v2f a0 = __builtin_nontemporal_load((const v2f*)(arow + ka0));
        v2f a1 = __builtin_nontemporal_load((const v2f*)(arow + ka1));
        v2f b0, b1;
        b0.x = W1[ka0 * F_HID + l16];       b0.y = W1[(ka0 + 1) * F_HID + l16];
        b1.x = W1[ka1 * F_HID + l16];       b1.y = W1[(ka1 + 1) * F_HID + l16];
        c0 = __builtin_amdgcn_wmma_f32_16x16x4_f32(false, a0, false, b0, (short)0, c0, false, false);
        c1 = __builtin_amdgcn_wmma_f32_16x16x4_f32(false, a1, false, b1, (short)0, c1, false, false);
    }
    for (; k < F_IN; k += 4) {                // tail (F_IN = 500 = 62*8 + 4)
        int ka = k + 2 * half;
        v2f a = __builtin_nontemporal_load((const v2f*)(arow + ka));
        v2f b;
        b.x = W1[ka * F_HID + l16];     b.y = W1[(ka + 1) * F_HID + l16];
        c0 = __builtin_amdgcn_wmma_f32_16x16x4_f32(false, a, false, b, (short)0, c0, false, false);
    }

    float* outp = h1 + (size_t)row0 * F_HID;
    if (row0 + 16 <= n) {                     // full tile: unguarded stores (common path)
#pragma unroll
        for (int r = 0; r < 8; ++r) {
            int rr = r + 8 * half;
            outp[rr * F_HID + l16] = c0[r] + c1[r];
        }
    } else {                                  // ragged tail tile (not hit for n=100000)
#pragma unroll
        for (int r = 0; r < 8; ++r) {
            int rr = r + 8 * half;
            if (row0 + rr < n) outp[rr * F_HID + l16] = c0[r] + c1[r];
        }
    }
}

// ---------------- scatter-add layer 1: agg1[dst] += h1[src]*norm ----------------
__global__ void scatter1_kernel(const int* __restrict__ src, const int* __restrict__ dst,
                                int E, int n, const float* __restrict__ dinv,
                                const float* __restrict__ h1, float* __restrict__ agg1) {
    int e = blockIdx.x * blockDim.x + threadIdx.x;
    if (e >= E + n) return;
    int s, d;
    if (e < E) { s = src[e]; d = dst[e]; } else { s = e - E; d = s; }   // self-loops
    float norm = dinv[s] * dinv[d];
    const float4* hv = (const float4*)(h1 + (size_t)s * F_HID);
    float* o = agg1 + (size_t)d * F_HID;
#pragma unroll
    for (int q = 0; q < 4; ++q) {
        float4 v = hv[q];
        atomicAdd(o + q * 4 + 0, v.x * norm);
        atomicAdd(o + q * 4 + 1, v.y * norm);
        atomicAdd(o + q * 4 + 2, v.z * norm);
        atomicAdd(o + q * 4 + 3, v.w * norm);
    }
}

// ---------------- bias + ReLU + tiny GEMM (16x3) per node ----------------
__global__ void layer2_kernel(const float* __restrict__ agg1, const float* __restrict__ b1,
                              const float* __restrict__ W2, float* __restrict__ h2, int n) {
    int i = blockIdx.x * blockDim.x + threadIdx.x;
    if (i >= n) return;
    float v[F_HID];
    const float4* a = (const float4*)(agg1 + (size_t)i * F_HID);
#pragma unroll
    for (int q = 0; q < 4; ++q) {
        float4 t = a[q];
        v[q * 4 + 0] = fmaxf(t.x + b1[q * 4 + 0], 0.0f);
        v[q * 4 + 1] = fmaxf(t.y + b1[q * 4 + 1], 0.0f);
        v[q * 4 + 2] = fmaxf(t.z + b1[q * 4 + 2], 0.0f);
        v[q * 4 + 3] = fmaxf(t.w + b1[q * 4 + 3], 0.0f);
    }
#pragma unroll
    for (int c = 0; c < F_OUT; ++c) {
        float acc = 0.0f;
#pragma unroll
        for (int f = 0; f < F_HID; ++f) acc += v[f] * W2[f * F_OUT + c];
        h2[(size_t)i * F_OUT + c] = acc;
    }
}

// ---------------- scatter-add layer 2 ----------------
__global__ void scatter2_kernel(const int* __restrict__ src, const int* __restrict__ dst,
                                int E, int n, const float* __restrict__ dinv,
                                const float* __restrict__ h2, float* __restrict__ agg2) {
    int e = blockIdx.x * blockDim.x + threadIdx.x;
    if (e >= E + n) return;
    int s, d;
    if (e < E) { s = src[e]; d = dst[e]; } else { s = e - E; d = s; }
    float norm = dinv[s] * dinv[d];
    const float* hp = h2 + (size_t)s * F_OUT;
    float* o = agg2 + (size_t)d * F_OUT;
#pragma unroll
    for (int c = 0; c < F_OUT; ++c) atomicAdd(o + c, hp[c] * norm);
}

// ---------------- + b2, log_softmax over 3 classes ----------------
__global__ void final_kernel(const float* __restrict__ agg2, const float* __restrict__ b2,
                             float* __restrict__ out, int n) {
    int i = blockIdx.x * blockDim.x + threadIdx.x;
    if (i >= n) return;
    float z0 = agg2[(size_t)i * F_OUT + 0] + b2[0];
    float z1 = agg2[(size_t)i * F_OUT + 1] + b2[1];
    float z2 = agg2[(size_t)i * F_OUT + 2] + b2[2];
    float m = fmaxf(z0, fmaxf(z1, z2));
    float lse = m + logf(expf(z0 - m) + expf(z1 - m) + expf(z2 - m));
    out[(size_t)i * F_OUT + 0] = z0 - lse;
    out[(size_t)i * F_OUT + 1] = z1 - lse;
    out[(size_t)i * F_OUT + 2] = z2 - lse;
}

extern "C" void kernel_launch(void* const* d_in, const int* in_sizes, int n_in,
                              void* d_out, int out_size, void* d_ws, size_t ws_size,
                              hipStream_t stream) {
    const float* x  = (const float*)d_in[0];
    const int* edge = (const int*)  d_in[1];   // [2, E] int32: row0=src, row1=dst
    const float* W1 = (const float*)d_in[2];
    const float* b1 = (const float*)d_in[3];
    const float* W2 = (const float*)d_in[4];
    const float* b2 = (const float*)d_in[5];
    float* out = (float*)d_out;

    int n = in_sizes[0] / F_IN;
    int E = in_sizes[1] / 2;
    const int* src = edge;
    const int* dst = edge + E;

    // Carve scratch from d_ws (256B-aligned regions), ~15.6 MB total.
    char* ws = (char*)d_ws;
    size_t off = 0;
    auto carve = [&](size_t bytes) {
        char* p = ws + off;
        off = (off + bytes + 255) & ~(size_t)255;
        return p;
    };
    float* dinv = (float*)carve((size_t)n * sizeof(float));            // deg -> deg^-1/2
    float* h1   = (float*)carve((size_t)n * F_HID * sizeof(float));
    float* agg1 = (float*)carve((size_t)n * F_HID * sizeof(float));
    float* h2   = (float*)carve((size_t)n * F_OUT * sizeof(float));
    float* agg2 = (float*)carve((size_t)n * F_OUT * sizeof(float));

    const int T = 256;
    int initN = n * F_HID;
    init_kernel  <<<(initN + T - 1) / T, T, 0, stream>>>(dinv, agg1, agg2, n);
    degree_kernel<<<(E + T - 1) / T,     T, 0, stream>>>(dst, E, dinv);
    rsqrt_kernel <<<(n + T - 1) / T,     T, 0, stream>>>(dinv, n);

    int ntiles = (n + 15) / 16;            // one wave32 per 16-row tile
    const int WPB = 8;                     // 8 waves / 256 threads per block
    gemm1_wmma_kernel<<<(ntiles + WPB - 1) / WPB, 32 * WPB, 0, stream>>>(x, W1, h1, n, ntiles);

    int Etot = E + n;
    scatter1_kernel<<<(Etot + T - 1) / T, T, 0, stream>>>(src, dst, E, n, dinv, h1, agg1);
    layer2_kernel  <<<(n + T - 1) / T,    T, 0, stream>>>(agg1, b1, W2, h2, n);
    scatter2_kernel<<<(Etot + T - 1) / T, T, 0, stream>>>(src, dst, E, n, dinv, h2, agg2);
    final_kernel   <<<(n + T - 1) / T,    T, 0, stream>>>(agg2, b2, out, n);
}